// GuidedFilter_52072183497293
// MI455X (gfx1250) — compile-verified
//
#include <hip/hip_runtime.h>
#include <stdint.h>

// Fused guided filter (r=4, eps=0.1) for 8x3x768x768 fp32 on gfx1250.
//
// One workgroup computes one 32x32 output tile of one (n,c) plane:
//   1) async-load 48x48 x/y halos into LDS (global_load_async_to_lds_b128,
//      per-lane predicated for image borders; LDS pre-zeroed => zero padding)
//   2) horizontal 9-tap box of {x, y, x*y, x*x} via WMMA:
//      out16x16 = T(16x24) x Band(24x16), Band is a 0/1 matrix built in regs
//      => 6 x v_wmma_f32_16x16x4_f32 per 16x16 tile per quantity
//   3) vertical 9-tap sums (scalar) + closed-form N + A,b into LDS
//   4) horizontal 9-tap box of {A, b} via WMMA (same band trick)
//   5) vertical sums of hA,hb, out = meanA*x + meanb, store to global
//
// ~64.7 KB LDS per workgroup -> ~4 workgroups resident per WGP (320 KB).
// Memory-bound: ~311 MB total traffic @ 23.3 TB/s ~ 13 us lower bound.

typedef float v2f __attribute__((ext_vector_type(2)));
typedef float v8f __attribute__((ext_vector_type(8)));

#define GF_W    768
#define GF_H    768
#define GF_TILE 32
#define XP      56    // x/y halo pitch (cols 0..47 data, 48..55 zero pad for WMMA reads)
#define HP      52    // h-buffer pitch (48 rows x cols 0..47 written)
#define AP      44    // A/b pitch (40x40 valid)
#define GF_EPS  0.1f

__device__ __forceinline__ int imax(int a, int b) { return a > b ? a : b; }
__device__ __forceinline__ int imin(int a, int b) { return a < b ? a : b; }

// Flat shared-aperture pointer: low 32 bits are the LDS byte offset
// (aperture: addr = {SHARED_BASE[63:32], lds_offset}).
__device__ __forceinline__ unsigned lds_off(const void* p) {
  return (unsigned)(uintptr_t)p;
}

// Banded B-matrix fragment (4x16 slice of the 24x16 9-tap band, K = k0..k0+3).
// f32 B layout (2 VGPRs): VGPR0 = rows {K=0 | K=2} for lanes {0-15 | 16-31},
// VGPR1 = rows {K=1 | K=3}; column N = lane&15.  B[j][n] = (j-n) in [0,8].
__device__ __forceinline__ v2f band_frag(int k0, int lo, int hi) {
  int j = k0 + 2 * hi;
  v2f b;
  b.x = ((unsigned)(j - lo)     <= 8u) ? 1.0f : 0.0f;
  b.y = ((unsigned)(j + 1 - lo) <= 8u) ? 1.0f : 0.0f;
  return b;
}

__global__ __launch_bounds__(256) void guided_filter_fused(
    const float* __restrict__ x, const float* __restrict__ y,
    float* __restrict__ out)
{
  // LDS arena: 16192 floats = 64768 bytes
  __shared__ float smem[16192];
  float* xh = smem;                 // [48][56] x halo
  float* yh = smem + 2688;          // [48][56] y halo (region reused for A,b)
  float* Aa = smem + 2688;          // [40][44] A   (after yh is dead)
  float* Bb = smem + 2688 + 1760;   // [40][44] b
  float* h0 = smem + 6208;          // [48][52] hbox(x)   -> later hbox(A)
  float* h1 = h0 + 2496;            // [48][52] hbox(y)   -> later hbox(b)
  float* h2 = h1 + 2496;            // [48][52] hbox(x*y)
  float* h3 = h2 + 2496;            // [48][52] hbox(x*x)

  const int t    = threadIdx.x;
  const int wave = t >> 5;          // 8 waves (wave32)
  const int lane = t & 31;
  const int lo   = lane & 15;
  const int hi   = lane >> 4;

  const int plane = blockIdx.z;
  const int gx0   = blockIdx.x * GF_TILE;
  const int gy0   = blockIdx.y * GF_TILE;
  const float* xp = x + (size_t)plane * (GF_W * GF_H);
  const float* yp = y + (size_t)plane * (GF_W * GF_H);

  // ---- 0) zero x/y halo regions (gives zero padding at image borders and
  //         keeps WMMA pad-column reads finite; 0 * NaN would poison sums)
  for (int i = t; i < 2688 + 3520; i += 256) smem[i] = 0.0f;
  __syncthreads();

  // ---- 1) async global->LDS halo loads: 48 rows x 12 float4 quads.
  //         Halo origin (gy0-8, gx0-8); quads are 16B-aligned and always
  //         fully in- or fully out-of-image (W%4==0, halo offset % 4 == 0).
  for (int i = t; i < 576; i += 256) {
    int r  = i / 12;
    int q  = i - r * 12;
    int gr = gy0 - 8 + r;
    int gc = gx0 - 8 + q * 4;
    if ((unsigned)gr < GF_H && (unsigned)gc < GF_W) {
      unsigned goff = (unsigned)(gr * GF_W + gc) * 4u;
      unsigned lx   = lds_off(&xh[r * XP + q * 4]);
      unsigned ly   = lds_off(&yh[r * XP + q * 4]);
      asm volatile("global_load_async_to_lds_b128 %0, %1, %2"
                   :: "v"(lx), "v"(goff), "s"(xp) : "memory");
      asm volatile("global_load_async_to_lds_b128 %0, %1, %2"
                   :: "v"(ly), "v"(goff), "s"(yp) : "memory");
    }
  }
  asm volatile("s_wait_asynccnt 0" ::: "memory");
  __syncthreads();

  // ---- 2) stage-1 horizontal box via WMMA: 3x3 output tiles over 48x48.
  //         A frag (16x4 f32): lane holds row m0+lo, cols n0+k0+2*hi..+1.
  for (int tile = wave; tile < 9; tile += 8) {
    const int m0 = (tile / 3) * 16;
    const int n0 = (tile % 3) * 16;
    v8f cx = {}, cy = {}, cxy = {}, cxx = {};
    for (int k0 = 0; k0 < 24; k0 += 4) {
      const int col = n0 + k0 + 2 * hi;
      v2f ax  = *(const v2f*)&xh[(m0 + lo) * XP + col];
      v2f ay  = *(const v2f*)&yh[(m0 + lo) * XP + col];
      v2f axy = ax * ay;                       // elementwise == fragment of x*y
      v2f axx = ax * ax;
      v2f bb  = band_frag(k0, lo, hi);
      cx  = __builtin_amdgcn_wmma_f32_16x16x4_f32(false, ax,  false, bb, (short)0, cx,  false, false);
      cy  = __builtin_amdgcn_wmma_f32_16x16x4_f32(false, ay,  false, bb, (short)0, cy,  false, false);
      cxy = __builtin_amdgcn_wmma_f32_16x16x4_f32(false, axy, false, bb, (short)0, cxy, false, false);
      cxx = __builtin_amdgcn_wmma_f32_16x16x4_f32(false, axx, false, bb, (short)0, cxx, false, false);
    }
    // C/D layout: VGPR v -> row m0 + v + 8*hi, col n0 + lo
    for (int v = 0; v < 8; ++v) {
      const int o = (m0 + v + 8 * hi) * HP + n0 + lo;
      h0[o] = cx[v]; h1[o] = cy[v]; h2[o] = cxy[v]; h3[o] = cxx[v];
    }
  }
  __syncthreads();

  // ---- 3) stage-1 vertical sums + A,b on the 40x40 (output+4) region.
  //         N computed in closed form == box(ones) with zero padding.
  for (int idx = t; idx < 1600; idx += 256) {
    const int r = idx / 40;
    const int c = idx - r * 40;
    float sx = 0.f, sy = 0.f, sxy = 0.f, sxx = 0.f;
    for (int i = 0; i < 9; ++i) {
      const int o = (r + i) * HP + c;
      sx += h0[o]; sy += h1[o]; sxy += h2[o]; sxx += h3[o];
    }
    const int gr = gy0 + r - 4;
    const int gc = gx0 + c - 4;
    float Av = 0.f, bv = 0.f;
    if ((unsigned)gr < GF_H && (unsigned)gc < GF_W) {
      const int ny = imin(gr + 4, GF_H - 1) - imax(gr - 4, 0) + 1;
      const int nx = imin(gc + 4, GF_W - 1) - imax(gc - 4, 0) + 1;
      const float Ninv = 1.0f / (float)(ny * nx);
      const float mx = sx * Ninv, my = sy * Ninv;
      const float cov = sxy * Ninv - mx * my;
      const float var = sxx * Ninv - mx * mx;
      Av = cov / (var + GF_EPS);
      bv = my - Av * mx;
    }
    Aa[r * AP + c] = Av;        // zero outside image = zero padding for box(A)
    Bb[r * AP + c] = bv;
  }
  __syncthreads();

  // ---- 4) stage-2 horizontal box of A,b via WMMA.
  //         Row tiles {0,16,24} (overlap rows 24..31: identical values),
  //         col tiles {0,16}; reads stay within the 40x40 valid region.
  if (wave < 6) {
    const int tm = wave >> 1;
    const int m0 = (tm == 2) ? 24 : tm * 16;
    const int n0 = (wave & 1) * 16;
    v8f ca = {}, cb = {};
    for (int k0 = 0; k0 < 24; k0 += 4) {
      const int col = n0 + k0 + 2 * hi;
      v2f aA = *(const v2f*)&Aa[(m0 + lo) * AP + col];
      v2f ab = *(const v2f*)&Bb[(m0 + lo) * AP + col];
      v2f bb = band_frag(k0, lo, hi);
      ca = __builtin_amdgcn_wmma_f32_16x16x4_f32(false, aA, false, bb, (short)0, ca, false, false);
      cb = __builtin_amdgcn_wmma_f32_16x16x4_f32(false, ab, false, bb, (short)0, cb, false, false);
    }
    for (int v = 0; v < 8; ++v) {
      const int o = (m0 + v + 8 * hi) * HP + n0 + lo;   // reuse h0/h1 as hA/hb
      h0[o] = ca[v];
      h1[o] = cb[v];
    }
  }
  __syncthreads();

  // ---- 5) stage-2 vertical sums, normalize, final blend, store.
  float* op = out + (size_t)plane * (GF_W * GF_H);
  for (int idx = t; idx < 1024; idx += 256) {
    const int r = idx >> 5;
    const int c = idx & 31;
    float sA = 0.f, sB = 0.f;
    for (int i = 0; i < 9; ++i) {
      const int o = (r + i) * HP + c;
      sA += h0[o]; sB += h1[o];
    }
    const int gr = gy0 + r;
    const int gc = gx0 + c;
    const int ny = imin(gr + 4, GF_H - 1) - imax(gr - 4, 0) + 1;
    const int nx = imin(gc + 4, GF_W - 1) - imax(gc - 4, 0) + 1;
    const float Ninv = 1.0f / (float)(ny * nx);
    const float xc = xh[(r + 8) * XP + (c + 8)];
    op[gr * GF_W + gc] = (sA * Ninv) * xc + (sB * Ninv);
  }
}

extern "C" void kernel_launch(void* const* d_in, const int* in_sizes, int n_in,
                              void* d_out, int out_size, void* d_ws, size_t ws_size,
                              hipStream_t stream) {
  (void)in_sizes; (void)n_in; (void)d_ws; (void)ws_size;
  const float* x = (const float*)d_in[0];
  const float* y = (const float*)d_in[1];
  float* out = (float*)d_out;
  const int planes = out_size / (GF_W * GF_H);   // 8*3 = 24
  dim3 grid(GF_W / GF_TILE, GF_H / GF_TILE, planes);
  guided_filter_fused<<<grid, 256, 0, stream>>>(x, y, out);
}